// GeometryConsistencyLoss_4217657885377
// MI455X (gfx1250) — compile-verified
//
#include <hip/hip_runtime.h>
#include <hip/hip_bf16.h>

typedef __attribute__((ext_vector_type(2))) float v2f;
typedef __attribute__((ext_vector_type(8))) float v8f;

#define B_N   512
#define D_DIM 512
#define HW    65536

// ---------------------------------------------------------------------------
// Kernel 1: per-image depth mean / std (ddof=1).  One 256-thread block per
// image; each thread streams 64 float4s (fully coalesced), LDS tree reduce.
// ---------------------------------------------------------------------------
__global__ void depth_stats_kernel(const float* __restrict__ depth,
                                   float* __restrict__ stats /* [B][2] */) {
    __shared__ float s_sum[256];
    __shared__ float s_sq[256];
    const int row = blockIdx.x;
    const int t   = threadIdx.x;
    const float4* p = (const float4*)(depth + (size_t)row * HW);
    float s = 0.f, q = 0.f;
#pragma unroll 4
    for (int i = 0; i < 64; ++i) {
        float4 v = p[i * 256 + t];
        s += v.x + v.y + v.z + v.w;
        q += v.x * v.x + v.y * v.y + v.z * v.z + v.w * v.w;
    }
    s_sum[t] = s;
    s_sq[t]  = q;
    __syncthreads();
    for (int off = 128; off > 0; off >>= 1) {
        if (t < off) { s_sum[t] += s_sum[t + off]; s_sq[t] += s_sq[t + off]; }
        __syncthreads();
    }
    if (t == 0) {
        float sum  = s_sum[0];
        float sq   = s_sq[0];
        float mean = sum / (float)HW;
        float var  = (sq - sum * mean) / (float)(HW - 1);  // ddof=1
        var = fmaxf(var, 0.f);
        stats[2 * row + 0] = mean;
        stats[2 * row + 1] = sqrtf(var);
    }
}

// ---------------------------------------------------------------------------
// Kernel 2: row squared-norms of features_2d / features_3d.
// One wave per row (8 rows per 256-thread block); grid.y selects dataset.
// ---------------------------------------------------------------------------
__global__ void rownorm_kernel(const float* __restrict__ X0,
                               const float* __restrict__ X1,
                               float* __restrict__ n0,
                               float* __restrict__ n1) {
    const int wave = threadIdx.x >> 5;
    const int lane = threadIdx.x & 31;
    const int row  = blockIdx.x * 8 + wave;
    const float* X = blockIdx.y ? X1 : X0;
    float* out     = blockIdx.y ? n1 : n0;
    const float4* p = (const float4*)(X + (size_t)row * D_DIM);
    float q = 0.f;
#pragma unroll
    for (int i = 0; i < 4; ++i) {
        float4 v = p[i * 32 + lane];
        q += v.x * v.x + v.y * v.y + v.z * v.z + v.w * v.w;
    }
    for (int off = 16; off > 0; off >>= 1) q += __shfl_xor(q, off, 32);
    if (lane == 0) out[row] = q;
}

// ---------------------------------------------------------------------------
// Kernel 3: WMMA gram tile + partial loss.
// One wave (32 threads) computes a 16x16 tile of G = X X^T via
// v_wmma_f32_16x16x4_f32 over K=512 (128 MACs), converts to Euclidean
// distance with the gram trick, subtracts the depth-feature distance,
// and wave-reduces the squared error into one partial per tile.
//
// ISA VGPR layouts (CDNA5 ISA):
//   A 16x4  f32: lane L, vgpr v -> A[M=L%16][K=2*(L/16)+v]
//   B  4x16 f32: lane L, vgpr v -> B[K=2*(L/16)+v][N=L%16]
//   C/D 16x16  : lane L, vgpr r -> D[M=r+8*(L/16)][N=L%16]
// For X X^T both A and B are the same per-lane row-gather of X.
// ---------------------------------------------------------------------------
__global__ void gram_loss_kernel(const float* __restrict__ F2,
                                 const float* __restrict__ F3,
                                 const float* __restrict__ n2,
                                 const float* __restrict__ n3,
                                 const float* __restrict__ stats,
                                 float* __restrict__ partials) {
    const float* X  = blockIdx.z ? F3 : F2;
    const float* x2 = blockIdx.z ? n3 : n2;
    const int rowBase = blockIdx.y * 16;
    const int colBase = blockIdx.x * 16;
    const int lane = threadIdx.x;
    const int half = lane >> 4;   // 0 or 1
    const int m    = lane & 15;

    const float* Ap = X + (size_t)(rowBase + m) * D_DIM + 2 * half;
    const float* Bp = X + (size_t)(colBase + m) * D_DIM + 2 * half;

    v8f acc = {};
#pragma unroll 4
    for (int k = 0; k < D_DIM; k += 4) {
        v2f a = *(const v2f*)(Ap + k);
        v2f b = *(const v2f*)(Bp + k);
        // (neg_a, A, neg_b, B, c_mod, C, reuse_a, reuse_b)
        acc = __builtin_amdgcn_wmma_f32_16x16x4_f32(
            false, a, false, b, (short)0, acc, false, false);
    }

    // Per-lane: column j fixed, 8 rows i across acc registers.
    const int j   = colBase + m;
    const float x2j = x2[j];
    const float mj  = stats[2 * j + 0];
    const float sj  = stats[2 * j + 1];

    float part = 0.f;
#pragma unroll
    for (int r = 0; r < 8; ++r) {
        const int i = rowBase + r + 8 * half;
        float g  = acc[r];
        float d2 = x2[i] + x2j - 2.f * g;
        d2 = fmaxf(d2, 0.f);
        float dist = d2 > 0.f ? sqrtf(d2) : 0.f;   // safe sqrt like reference
        float dm = stats[2 * i + 0] - mj;
        float ds = stats[2 * i + 1] - sj;
        float dd2 = dm * dm + ds * ds;
        float dd  = dd2 > 0.f ? sqrtf(dd2) : 0.f;
        float e = dist - dd;
        part += e * e;
    }

    for (int off = 16; off > 0; off >>= 1) part += __shfl_xor(part, off, 32);
    if (lane == 0) {
        const int tile = ((int)blockIdx.z * 32 + (int)blockIdx.y) * 32 + (int)blockIdx.x;
        partials[tile] = part;
    }
}

// ---------------------------------------------------------------------------
// Kernel 4: deterministic final reduction of 2048 tile partials + scale.
// loss = (sum2d + sum3d) / (2 * B^2)   (LAMBDA_WEIGHT = 1)
// ---------------------------------------------------------------------------
__global__ void final_reduce_kernel(const float* __restrict__ partials,
                                    float* __restrict__ out) {
    __shared__ float s[256];
    const int t = threadIdx.x;
    float v = 0.f;
    for (int i = t; i < 2048; i += 256) v += partials[i];
    s[t] = v;
    __syncthreads();
    for (int off = 128; off > 0; off >>= 1) {
        if (t < off) s[t] += s[t + off];
        __syncthreads();
    }
    if (t == 0) out[0] = s[0] * (1.0f / (2.0f * (float)B_N * (float)B_N));
}

extern "C" void kernel_launch(void* const* d_in, const int* in_sizes, int n_in,
                              void* d_out, int out_size, void* d_ws, size_t ws_size,
                              hipStream_t stream) {
    const float* features_2d = (const float*)d_in[0];  // [512, 512]
    const float* features_3d = (const float*)d_in[1];  // [512, 512]
    const float* depth_maps  = (const float*)d_in[2];  // [512, 1, 256, 256]
    float* out = (float*)d_out;

    // Workspace layout (floats): stats[1024] | n2[512] | n3[512] | partials[2048]
    float* stats    = (float*)d_ws;
    float* n2       = stats + 2 * B_N;
    float* n3       = n2 + B_N;
    float* partials = n3 + B_N;

    // 1) depth mean/std per image
    depth_stats_kernel<<<B_N, 256, 0, stream>>>(depth_maps, stats);

    // 2) row squared-norms for both feature sets (512 rows each, 8 rows/block)
    rownorm_kernel<<<dim3(B_N / 8, 2), 256, 0, stream>>>(features_2d, features_3d, n2, n3);

    // 3) WMMA gram tiles + per-tile partial loss (32x32 tiles x 2 datasets)
    gram_loss_kernel<<<dim3(32, 32, 2), 32, 0, stream>>>(
        features_2d, features_3d, n2, n3, stats, partials);

    // 4) deterministic final reduction
    final_reduce_kernel<<<1, 256, 0, stream>>>(partials, out);
}